// VQC_23398981829257
// MI455X (gfx1250) — compile-verified
//
#include <hip/hip_runtime.h>
#include <hip/hip_bf16.h>

// VQC, 4 qubits, DIM=16, 3 layers, BATCH=524288.
//
// Decomposition:
//   state_b = U * s_b, U = fixed 16x16 complex unitary from weights only,
//   s_b[k]  = prod_i (bit_i(k) ? sin(x_i/2) : cos(x_i/2))   (real product state)
//   out[b,w]= sum_k Z_w[k] * |(U s_b)[k]|^2
//
// Kernel 1: build U columns (16 lanes, registers only), store Ur/Ui row-major.
// Kernel 2: per 16-batch tile: F^T = Ur*S^T, G^T = Ui*S^T via 8x
//           V_WMMA_F32_16X16X4_F32 (A = U, wave-invariant; B = product states).
//           probs^T lands with batch row per lane and basis index k across the
//           8 D VGPRs (+ lane-group bit3), so the Z-signed sums are an
//           in-register 8-point partial WHT + ONE xor-16 shuffle per output.
//           Output row stored as a single coalesced b128.

typedef __attribute__((ext_vector_type(2))) float v2f;
typedef __attribute__((ext_vector_type(4))) float v4f;
typedef __attribute__((ext_vector_type(8))) float v8f;

#define N_LAYERS 3

// ---------------------------------------------------------------------------
// Kernel 1: simulate the weight-only part of the circuit on 16 basis vectors.
// Lane j owns column j of U. Writes A-operand layout: AUr[k*16+j] = Ur[k][j].
// ---------------------------------------------------------------------------
__global__ void vqc_build_u(const float* __restrict__ w, float* __restrict__ ws) {
    const int j = threadIdx.x;
    if (j >= 16) return;

    float re[16], im[16];
#pragma unroll
    for (int k = 0; k < 16; ++k) { re[k] = (k == j) ? 1.0f : 0.0f; im[k] = 0.0f; }

    for (int layer = 0; layer < N_LAYERS; ++layer) {
        // CNOT ring: control i -> target (i+1)%4 ; new[k] = old[k ^ (cbit<<tb)]
#pragma unroll
        for (int i = 0; i < 4; ++i) {
            const int cb = 3 - i;
            const int tb = 3 - ((i + 1) & 3);
            float tr[16], ti[16];
#pragma unroll
            for (int k = 0; k < 16; ++k) {
                int src = k ^ (((k >> cb) & 1) << tb);
                tr[k] = re[src]; ti[k] = im[src];
            }
#pragma unroll
            for (int k = 0; k < 16; ++k) { re[k] = tr[k]; im[k] = ti[k]; }
        }
        // Per-qubit RY(w[l,i,0]) then RZ(w[l,i,1])
#pragma unroll
        for (int i = 0; i < 4; ++i) {
            const int b = 3 - i;
            const float thy = w[(layer * 4 + i) * 2 + 0] * 0.5f;
            const float thz = w[(layer * 4 + i) * 2 + 1] * 0.5f;
            const float cy = cosf(thy), sy = sinf(thy);
#pragma unroll
            for (int k = 0; k < 16; ++k) {
                if (((k >> b) & 1) == 0) {
                    const int k1 = k | (1 << b);
                    float r0 = re[k], i0 = im[k], r1 = re[k1], i1 = im[k1];
                    re[k]  = cy * r0 - sy * r1;  im[k]  = cy * i0 - sy * i1;
                    re[k1] = sy * r0 + cy * r1;  im[k1] = sy * i0 + cy * i1;
                }
            }
            const float cz = cosf(thz), sz = sinf(thz);
#pragma unroll
            for (int k = 0; k < 16; ++k) {
                float r = re[k], q = im[k];
                if (((k >> b) & 1) == 0) { re[k] = cz * r + sz * q; im[k] = cz * q - sz * r; }
                else                     { re[k] = cz * r - sz * q; im[k] = cz * q + sz * r; }
            }
        }
    }

    float* AUr = ws;         // 16x16 f32 row-major: AUr[k][j] = Ur[k][j]
    float* AUi = ws + 256;   // 16x16 f32 row-major: AUi[k][j] = Ui[k][j]
#pragma unroll
    for (int k = 0; k < 16; ++k) {
        AUr[k * 16 + j] = re[k];
        AUi[k * 16 + j] = im[k];
    }
}

// ---------------------------------------------------------------------------
// Kernel 2: one 16-row batch tile per wave iteration.
// ---------------------------------------------------------------------------
__global__ void vqc_batch(const float* __restrict__ x,
                          const float* __restrict__ ws_u,
                          float* __restrict__ out, int ntiles) {
    const int lane = threadIdx.x & 31;
    const int ln   = lane & 15;      // batch row within tile (B's N, D's N)
    const int h    = lane >> 4;      // K lane-group (0/1); also bit3 of D's M
    const int wave   = blockIdx.x * (blockDim.x >> 5) + (threadIdx.x >> 5);
    const int nwaves = gridDim.x * (blockDim.x >> 5);

    // Wave-invariant A operands (U rows): for K-chunk t,
    // VGPR0 = A[M=ln][4t+2h], VGPR1 = A[M=ln][4t+2h+1].
    const float* AUr = ws_u;
    const float* AUi = ws_u + 256;
    v2f ar[4], ai[4];
#pragma unroll
    for (int t = 0; t < 4; ++t) {
        const int k0 = 4 * t + 2 * h;
        ar[t][0] = AUr[ln * 16 + k0];   ar[t][1] = AUr[ln * 16 + k0 + 1];
        ai[t][0] = AUi[ln * 16 + k0];   ai[t][1] = AUi[ln * 16 + k0 + 1];
    }

    const v4f* __restrict__ x4   = (const v4f*)x;
    v4f*       __restrict__ out4 = (v4f*)out;

    for (int tile = wave; tile < ntiles; tile += nwaves) {
        const int row = tile * 16 + ln;            // this lane's batch row
        // product-state factors from the 4 data angles (single b128 load)
        const v4f xv = x4[row];
        const float a0 = xv[0] * 0.5f, a1 = xv[1] * 0.5f;
        const float a2 = xv[2] * 0.5f, a3 = xv[3] * 0.5f;
        const float c0 = __cosf(a0), s0 = __sinf(a0);
        const float c1 = __cosf(a1), s1 = __sinf(a1);
        const float c2 = __cosf(a2), s2 = __sinf(a2);
        const float c3 = __cosf(a3), s3 = __sinf(a3);
        const float t0 = c0 * c1, t1 = c0 * s1, t2 = s0 * c1, t3 = s0 * s1;
        // B operand: lane feeds s_row components {4t+2h, 4t+2h+1}:
        //   s[4t+2h] = tvals[t]*(h?s2:c2)*c3 ; s[4t+2h+1] = tvals[t]*(h?s2:c2)*s3
        const float w2 = h ? s2 : c2;
        const float b0 = t0 * w2, b1 = t1 * w2, b2 = t2 * w2, b3 = t3 * w2;

        // F^T = Ur * S^T, G^T = Ui * S^T (K=16 as 4 chunks of 4)
        v8f f = {}, g = {};
        {
            v2f b;
            b[0] = b0 * c3; b[1] = b0 * s3;
            f = __builtin_amdgcn_wmma_f32_16x16x4_f32(false, ar[0], false, b, (short)0, f, false, false);
            g = __builtin_amdgcn_wmma_f32_16x16x4_f32(false, ai[0], false, b, (short)0, g, false, false);
            b[0] = b1 * c3; b[1] = b1 * s3;
            f = __builtin_amdgcn_wmma_f32_16x16x4_f32(false, ar[1], false, b, (short)0, f, false, false);
            g = __builtin_amdgcn_wmma_f32_16x16x4_f32(false, ai[1], false, b, (short)0, g, false, false);
            b[0] = b2 * c3; b[1] = b2 * s3;
            f = __builtin_amdgcn_wmma_f32_16x16x4_f32(false, ar[2], false, b, (short)0, f, false, false);
            g = __builtin_amdgcn_wmma_f32_16x16x4_f32(false, ai[2], false, b, (short)0, g, false, false);
            b[0] = b3 * c3; b[1] = b3 * s3;
            f = __builtin_amdgcn_wmma_f32_16x16x4_f32(false, ar[3], false, b, (short)0, f, false, false);
            g = __builtin_amdgcn_wmma_f32_16x16x4_f32(false, ai[3], false, b, (short)0, g, false, false);
        }

        // probs^T in D layout: VGPR r, lane (ln,h) -> probs[m=ln][k=r+8h]
        const float p0 = f[0] * f[0] + g[0] * g[0];
        const float p1 = f[1] * f[1] + g[1] * g[1];
        const float p2 = f[2] * f[2] + g[2] * g[2];
        const float p3 = f[3] * f[3] + g[3] * g[3];
        const float p4 = f[4] * f[4] + g[4] * g[4];
        const float p5 = f[5] * f[5] + g[5] * g[5];
        const float p6 = f[6] * f[6] + g[6] * g[6];
        const float p7 = f[7] * f[7] + g[7] * g[7];

        // In-register partial WHT over k-bits 2..0 (signs compile-time per VGPR):
        const float q01 = p0 + p1, q23 = p2 + p3, q45 = p4 + p5, q67 = p6 + p7;
        const float e0 = q01 + q23, e1 = q45 + q67;
        const float T = e0 + e1;                               // all +        (w=0 partial)
        const float A = e0 - e1;                               // sign bit2(k) (w=1)
        const float B = (q01 - q23) + (q45 - q67);             // sign bit1(k) (w=2)
        const float d01 = p0 - p1, d23 = p2 - p3, d45 = p4 - p5, d67 = p6 - p7;
        const float C = (d01 + d23) + (d45 + d67);             // sign bit0(k) (w=3)

        // bit3(k) == lane group h: one xor-16 combine per observable
        const float Tp = __shfl_xor(T, 16, 32);
        const float Ap = __shfl_xor(A, 16, 32);
        const float Bp = __shfl_xor(B, 16, 32);
        const float Cp = __shfl_xor(C, 16, 32);

        if (h == 0) {
            v4f o;
            o[0] = T - Tp;   // w=0: +(k<8) -(k>=8)
            o[1] = A + Ap;   // w=1
            o[2] = B + Bp;   // w=2
            o[3] = C + Cp;   // w=3
            out4[row] = o;   // coalesced b128 store of the full output row
        }
    }
}

extern "C" void kernel_launch(void* const* d_in, const int* in_sizes, int n_in,
                              void* d_out, int out_size, void* d_ws, size_t ws_size,
                              hipStream_t stream) {
    const float* x  = (const float*)d_in[0];   // (BATCH, 4) f32
    const float* w  = (const float*)d_in[1];   // (3, 4, 2) f32
    float* out = (float*)d_out;                // (BATCH, 4) f32
    float* ws  = (float*)d_ws;                 // 512 floats used (Ur/Ui row-major)

    const int batch  = in_sizes[0] / 4;
    const int ntiles = batch / 16;

    vqc_build_u<<<1, 32, 0, stream>>>(w, ws);

    const int threads = 256;                   // 8 waves/block
    const int blocks  = 1024;                  // 8192 waves; ~4 tiles per wave
    vqc_batch<<<blocks, threads, 0, stream>>>(x, ws, out, ntiles);
}